// PDE_SSM_Block_41635412967880
// MI455X (gfx1250) — compile-verified
//
#include <hip/hip_runtime.h>

typedef float v2f __attribute__((ext_vector_type(2)));
typedef float v8f __attribute__((ext_vector_type(8)));

#define D_IN 128
#define D_S  8
#define RNK  4
#define HW   4096          // 64*64
#define POS_TILE 32
#define THREADS 512

__device__ __forceinline__ float fast_sigmoid(float v) {
    return 1.0f / (1.0f + __expf(-v));
}
__device__ __forceinline__ float fast_tanh(float v) {
    // tanh(x) = 1 - 2/(1+exp(2x)) -> single v_exp_f32 + v_rcp_f32
    return 1.0f - 2.0f / (1.0f + __expf(2.0f * v));
}

__global__ __launch_bounds__(THREADS)
void pde_ssm_kernel(const float* __restrict__ x,
                    const float* __restrict__ h_prev,
                    const float* __restrict__ x_proj_w,
                    const float* __restrict__ x_proj_b,
                    const float* __restrict__ delta_lift_w,
                    const float* __restrict__ delta_lift_b,
                    const float* __restrict__ gate_B_w,
                    const float* __restrict__ gate_B_b,
                    const float* __restrict__ gate_C_w,
                    const float* __restrict__ gate_C_b,
                    const float* __restrict__ B_factor,
                    const float* __restrict__ B_out,
                    const float* __restrict__ C_factor,
                    const float* __restrict__ C_out,
                    const float* __restrict__ A_log,
                    const float* __restrict__ D_skip,
                    const float* __restrict__ dyn_scale_p,
                    float* __restrict__ y_out,
                    float* __restrict__ h_out)
{
    __shared__ alignas(16) float xs [D_IN * POS_TILE];   // x tile [d][n]
    __shared__ alignas(16) float gBs[D_IN * POS_TILE];   // sigmoid(gate_B)
    __shared__ alignas(16) float gCs[D_IN * POS_TILE];   // sigmoid(gate_C)
    __shared__ alignas(16) float dls[D_IN * POS_TILE];   // delta
    __shared__ alignas(16) float ps [12 * POS_TILE];     // x_proj params [j][n]
    __shared__ alignas(16) float Aneg[D_IN * D_S];       // -exp(A_log)
    __shared__ alignas(16) float Bo_s[RNK * D_S];
    __shared__ alignas(16) float Co_s[RNK * D_S];

    const int t  = threadIdx.x;
    const int b  = blockIdx.x >> 7;                 // 128 position-tiles per batch
    const int p0 = (blockIdx.x & 127) * POS_TILE;

    const float* xb  = x      + (size_t)b * D_IN * HW        + p0;
    const float* hb  = h_prev + (size_t)b * D_IN * D_S * HW  + p0;
    float*       yb  = y_out  + (size_t)b * D_IN * HW        + p0;
    float*       hob = h_out  + (size_t)b * D_IN * D_S * HW  + p0;

    // ---- Phase 0: stage x tile + A into LDS (128B-coalesced) --------------
    #pragma unroll
    for (int i = 0; i < 8; ++i) {
        int e = t + THREADS * i;                    // e = d*32 + n
        int d = e >> 5, n = e & 31;
        xs[e] = xb[d * HW + n];
    }
    #pragma unroll
    for (int i = 0; i < 2; ++i) {
        int e = t + THREADS * i;
        Aneg[e] = -__expf(A_log[e]);
    }
    if (t < RNK * D_S) { Bo_s[t] = B_out[t]; Co_s[t] = C_out[t]; }
    __syncthreads();

    // ---- Phase 1: gate GEMMs on the matrix pipes (fp32 WMMA, exact) -------
    // 16 waves: gate = w/8, rows 32*(w%4)..+32, cols 16*((w%8)/4)..+16
    {
        const int w    = t >> 5;
        const int lane = t & 31;
        const int nl   = lane & 15;
        const int kh   = lane >> 4;                 // K half-select
        const int sub  = w & 7;
        const int m0   = (sub & 3) * 32;
        const int n0   = (sub >> 2) * 16;
        const int n    = n0 + nl;
        const float* Wg   = (w < 8) ? gate_B_w : gate_C_w;
        const float* bg   = (w < 8) ? gate_B_b : gate_C_b;
        float*       gout = (w < 8) ? gBs      : gCs;

        v8f acc0 = {};
        v8f acc1 = {};
        #pragma unroll 4
        for (int kk = 0; kk < 32; ++kk) {
            const int kbase = kk * 4 + kh * 2;      // lane's K pair within 4-wide step
            v2f a0 = *(const v2f*)&Wg[(size_t)(m0 + nl)      * D_IN + kbase];
            v2f a1 = *(const v2f*)&Wg[(size_t)(m0 + 16 + nl) * D_IN + kbase];
            v2f bf;
            bf.x = xs[(kbase    ) * POS_TILE + n];
            bf.y = xs[(kbase + 1) * POS_TILE + n];
            acc0 = __builtin_amdgcn_wmma_f32_16x16x4_f32(false, a0, false, bf,
                                                         (short)0, acc0, false, false);
            acc1 = __builtin_amdgcn_wmma_f32_16x16x4_f32(false, a1, false, bf,
                                                         (short)0, acc1, false, false);
        }
        #pragma unroll
        for (int j = 0; j < 8; ++j) {
            int m = m0 + j + 8 * kh;                // C/D layout: VGPR j -> M = j + 8*(lane/16)
            gout[m * POS_TILE + n]        = fast_sigmoid(acc0[j] + bg[m]);
            gout[(m + 16) * POS_TILE + n] = fast_sigmoid(acc1[j] + bg[m + 16]);
        }
    }

    // ---- Phase 2: x_proj (12 x 32), tiny -> VALU --------------------------
    if (t < 12 * POS_TILE) {                        // waves 0..11, no divergence
        int j = t >> 5, n = t & 31;
        float acc = x_proj_b[j];
        #pragma unroll 8
        for (int k = 0; k < D_IN; ++k)
            acc += x_proj_w[j * D_IN + k] * xs[k * POS_TILE + n];
        ps[j * POS_TILE + n] = acc;
    }
    __syncthreads();

    // ---- Phase 3: delta = sigmoid(delta_lift_w @ delta_r + b) -------------
    #pragma unroll
    for (int i = 0; i < 8; ++i) {
        int e = t + THREADS * i;
        int d = e >> 5, n = e & 31;
        float acc = delta_lift_b[d];
        #pragma unroll
        for (int r = 0; r < RNK; ++r)
            acc += delta_lift_w[d * RNK + r] * ps[r * POS_TILE + n];
        dls[e] = fast_sigmoid(acc);
    }
    __syncthreads();

    // ---- Phase 4: fused state update, float4 over positions ---------------
    // thread -> 4 consecutive positions; 8 lanes cover a full 128B row segment
    const float scale = dyn_scale_p[0];
    const int nq = (t & 7) * 4;                     // position quad
    const int g  = t >> 3;                          // 64 groups, 2 channels each
    float Br[RNK][4], Cr[RNK][4];
    #pragma unroll
    for (int r = 0; r < RNK; ++r) {
        float4 tb = *(const float4*)&ps[(RNK     + r) * POS_TILE + nq];
        float4 tc = *(const float4*)&ps[(2 * RNK + r) * POS_TILE + nq];
        Br[r][0] = tb.x; Br[r][1] = tb.y; Br[r][2] = tb.z; Br[r][3] = tb.w;
        Cr[r][0] = tc.x; Cr[r][1] = tc.y; Cr[r][2] = tc.z; Cr[r][3] = tc.w;
    }
    #pragma unroll
    for (int dd = 0; dd < 2; ++dd) {
        const int d = g * 2 + dd;
        float4 t4;
        t4 = *(const float4*)&xs [d * POS_TILE + nq];
        const float xv[4]  = {t4.x, t4.y, t4.z, t4.w};
        t4 = *(const float4*)&dls[d * POS_TILE + nq];
        const float del[4] = {t4.x, t4.y, t4.z, t4.w};
        t4 = *(const float4*)&gBs[d * POS_TILE + nq];
        const float gb[4]  = {t4.x, t4.y, t4.z, t4.w};
        t4 = *(const float4*)&gCs[d * POS_TILE + nq];
        const float gc[4]  = {t4.x, t4.y, t4.z, t4.w};

        const float4 bf4 = *(const float4*)&B_factor[d * RNK];
        const float4 cf4 = *(const float4*)&C_factor[d * RNK];
        const float Bf[4] = {bf4.x, bf4.y, bf4.z, bf4.w};
        const float Cf[4] = {cf4.x, cf4.y, cf4.z, cf4.w};

        float vB[RNK][4], vC[RNK][4];
        #pragma unroll
        for (int r = 0; r < RNK; ++r)
            #pragma unroll
            for (int q = 0; q < 4; ++q) {
                vB[r][q] = Bf[r] * Br[r][q];
                vC[r][q] = Cf[r] * Cr[r][q];
            }

        float yacc[4] = {0.f, 0.f, 0.f, 0.f};
        #pragma unroll
        for (int s = 0; s < D_S; ++s) {
            float baseB = 0.f, baseC = 0.f;
            float dynB[4] = {0,0,0,0}, dynC[4] = {0,0,0,0};
            #pragma unroll
            for (int r = 0; r < RNK; ++r) {
                const float bo = Bo_s[r * D_S + s];
                const float co = Co_s[r * D_S + s];
                baseB += Bf[r] * bo;
                baseC += Cf[r] * co;
                #pragma unroll
                for (int q = 0; q < 4; ++q) {
                    dynB[q] += vB[r][q] * bo;
                    dynC[q] += vC[r][q] * co;
                }
            }
            const float an = Aneg[d * D_S + s];
            const size_t off = (size_t)(d * D_S + s) * HW + nq;
            const float4 h4 = *(const float4*)&hb[off];
            const float hp[4] = {h4.x, h4.y, h4.z, h4.w};
            float hn[4];
            #pragma unroll
            for (int q = 0; q < 4; ++q) {
                const float Beff = (baseB + scale * fast_tanh(dynB[q])) * gb[q];
                const float Ceff = (baseC + scale * fast_tanh(dynC[q])) * gc[q];
                const float tm   = __expf(an * del[q]);
                hn[q] = tm * hp[q] + del[q] * xv[q] * Beff;
                yacc[q] += hn[q] * Ceff;
            }
            float4 o4; o4.x = hn[0]; o4.y = hn[1]; o4.z = hn[2]; o4.w = hn[3];
            *(float4*)&hob[off] = o4;
        }
        const float dsk = D_skip[d];
        float4 y4;
        y4.x = yacc[0] + xv[0] * dsk;
        y4.y = yacc[1] + xv[1] * dsk;
        y4.z = yacc[2] + xv[2] * dsk;
        y4.w = yacc[3] + xv[3] * dsk;
        *(float4*)&yb[d * HW + nq] = y4;
    }
}

extern "C" void kernel_launch(void* const* d_in, const int* in_sizes, int n_in,
                              void* d_out, int out_size, void* d_ws, size_t ws_size,
                              hipStream_t stream) {
    const float* x            = (const float*)d_in[0];
    const float* h_prev       = (const float*)d_in[1];
    const float* x_proj_w     = (const float*)d_in[2];
    const float* x_proj_b     = (const float*)d_in[3];
    const float* delta_lift_w = (const float*)d_in[4];
    const float* delta_lift_b = (const float*)d_in[5];
    const float* gate_B_w     = (const float*)d_in[6];
    const float* gate_B_b     = (const float*)d_in[7];
    const float* gate_C_w     = (const float*)d_in[8];
    const float* gate_C_b     = (const float*)d_in[9];
    const float* B_factor     = (const float*)d_in[10];
    const float* B_out        = (const float*)d_in[11];
    const float* C_factor     = (const float*)d_in[12];
    const float* C_out        = (const float*)d_in[13];
    const float* A_log        = (const float*)d_in[14];
    const float* D_skip       = (const float*)d_in[15];
    const float* dyn_scale    = (const float*)d_in[16];

    float* y_out = (float*)d_out;                       // (2,128,64,64)
    float* h_out = y_out + (size_t)2 * D_IN * HW;       // (2,1024,64,64)

    const int blocks = 2 * (HW / POS_TILE);             // 256
    pde_ssm_kernel<<<blocks, THREADS, 0, stream>>>(
        x, h_prev, x_proj_w, x_proj_b, delta_lift_w, delta_lift_b,
        gate_B_w, gate_B_b, gate_C_w, gate_C_b,
        B_factor, B_out, C_factor, C_out, A_log, D_skip, dyn_scale,
        y_out, h_out);
}